// ContrastLoss_52948356825681
// MI455X (gfx1250) — compile-verified
//
#include <hip/hip_runtime.h>
#include <math.h>

// Problem constants (from the reference file)
#define TEMP      0.07f
#define BASE_TEMP 0.07f
#define B_DIM     16
#define D_DIM     128
#define L_DIM     20000

typedef float v2f __attribute__((ext_vector_type(2)));
typedef float v8f __attribute__((ext_vector_type(8)));

// ---------------------------------------------------------------------------
// Kernel 1: gather anchor matrix A[N][D] from feats via sample_idx.
// anchor row n = v*C + c  ->  feats_flat[sample_idx[c][v]][d]
// feats_flat[(b*L + pos)][d] == feats[b][d][pos]
// ---------------------------------------------------------------------------
__global__ void gather_anchor(const float* __restrict__ feats,
                              const int*   __restrict__ sample_idx,
                              float*       __restrict__ A,
                              int C, int V) {
  const int n = blockIdx.x;        // anchor row
  const int d = threadIdx.x;       // 0..127
  const int c = n % C;
  const int v = n / C;
  const int flat = sample_idx[c * V + v];
  const int b   = flat / L_DIM;
  const int pos = flat % L_DIM;
  A[(size_t)n * D_DIM + d] =
      feats[((size_t)b * D_DIM + d) * L_DIM + pos];
}

// ---------------------------------------------------------------------------
// Kernel 2: logits = (A * A^T) / TEMP using V_WMMA_F32_16X16X4_F32.
// Block = (32,4): 4 waves; each wave computes one 16x64 output strip
// (4 adjacent 16x16 tiles sharing the A fragment).
//
// Software-pipelined: fragments for k-step i+1 are loaded BEFORE the 4 WMMAs
// that consume step i, so global-load latency overlaps matrix ops instead of
// a full s_wait_loadcnt before every v_wmma.
//
// Per ISA layouts (wave32):
//   A frag 16x4:  lane%16 = M row, lane/16 selects K pair; float2 = {K, K+1}
//   B frag 4x16:  symmetric (B[k][n] = A[n][k] since we do A*A^T)
//   D 16x16 f32:  8 VGPRs; VGPR r holds M=r (lanes 0-15) / M=r+8 (lanes 16-31),
//                 N = lane%16.
// ---------------------------------------------------------------------------
__global__ void wmma_gram(const float* __restrict__ A,
                          float*       __restrict__ logits,
                          int N) {
  const int lane     = threadIdx.x;                 // 0..31
  const int wave     = threadIdx.y;                 // 0..3
  const int colGroup = blockIdx.x * 4 + wave;       // each covers 64 columns
  const int rowTile  = blockIdx.y;                  // each covers 16 rows
  if (colGroup * 64 >= N) return;

  const int half = lane >> 4;                       // 0 or 1
  const int lrow = lane & 15;
  const int mRow = rowTile * 16 + lrow;
  const int koff = 2 * half;                        // this lane's K pair base

  const float* aBase = A + (size_t)mRow * D_DIM + koff;
  const float* bBase0 = A + (size_t)(colGroup * 64 +  0 + lrow) * D_DIM + koff;
  const float* bBase1 = A + (size_t)(colGroup * 64 + 16 + lrow) * D_DIM + koff;
  const float* bBase2 = A + (size_t)(colGroup * 64 + 32 + lrow) * D_DIM + koff;
  const float* bBase3 = A + (size_t)(colGroup * 64 + 48 + lrow) * D_DIM + koff;

  v8f acc0 = {}, acc1 = {}, acc2 = {}, acc3 = {};

  // ---- prologue: load k-step 0 fragments ----
  v2f a  = *(const v2f*)(aBase);
  v2f b0 = *(const v2f*)(bBase0);
  v2f b1 = *(const v2f*)(bBase1);
  v2f b2 = *(const v2f*)(bBase2);
  v2f b3 = *(const v2f*)(bBase3);

#pragma unroll 4
  for (int k = 4; k < D_DIM; k += 4) {
    // issue next step's loads first (overlap with the WMMAs below)
    const v2f an  = *(const v2f*)(aBase  + k);
    const v2f bn0 = *(const v2f*)(bBase0 + k);
    const v2f bn1 = *(const v2f*)(bBase1 + k);
    const v2f bn2 = *(const v2f*)(bBase2 + k);
    const v2f bn3 = *(const v2f*)(bBase3 + k);

    acc0 = __builtin_amdgcn_wmma_f32_16x16x4_f32(false, a, false, b0,
                                                 (short)0, acc0, false, false);
    acc1 = __builtin_amdgcn_wmma_f32_16x16x4_f32(false, a, false, b1,
                                                 (short)0, acc1, false, false);
    acc2 = __builtin_amdgcn_wmma_f32_16x16x4_f32(false, a, false, b2,
                                                 (short)0, acc2, false, false);
    acc3 = __builtin_amdgcn_wmma_f32_16x16x4_f32(false, a, false, b3,
                                                 (short)0, acc3, false, false);

    a = an; b0 = bn0; b1 = bn1; b2 = bn2; b3 = bn3;
  }

  // ---- epilogue: last k-step ----
  acc0 = __builtin_amdgcn_wmma_f32_16x16x4_f32(false, a, false, b0,
                                               (short)0, acc0, false, false);
  acc1 = __builtin_amdgcn_wmma_f32_16x16x4_f32(false, a, false, b1,
                                               (short)0, acc1, false, false);
  acc2 = __builtin_amdgcn_wmma_f32_16x16x4_f32(false, a, false, b2,
                                               (short)0, acc2, false, false);
  acc3 = __builtin_amdgcn_wmma_f32_16x16x4_f32(false, a, false, b3,
                                               (short)0, acc3, false, false);

  const float invT = 1.0f / TEMP;
  const int colB = colGroup * 64 + lrow;
  const int rowB = rowTile * 16 + 8 * half;
#pragma unroll
  for (int r = 0; r < 8; ++r) {
    const size_t rb = (size_t)(rowB + r) * N;
    logits[rb + colB +  0] = acc0[r] * invT;
    logits[rb + colB + 16] = acc1[r] * invT;
    logits[rb + colB + 32] = acc2[r] * invT;
    logits[rb + colB + 48] = acc3[r] * invT;
  }
}

// ---------------------------------------------------------------------------
// Kernel 3: per-row contrastive reduction. One block (256 threads) per row i:
//   m      = max_j logits[i][j]
//   s_neg  = sum_j [y(i)!=y(j)] exp(logits[i][j]-m)
//   ps,cnt = sum over positives (same class, j!=i) of
//            (lv - log(exp(lv)+s_neg)), count
//   rowvals[i] = -(T/BT) * ps/cnt
// class of row j is y_anchor[j % C]   (mask = tile(mask_small,(V,V)))
// ---------------------------------------------------------------------------
__global__ void row_reduce(const float* __restrict__ logits,
                           const float* __restrict__ y,
                           float*       __restrict__ rowvals,
                           int N, int C) {
  const int i   = blockIdx.x;
  const int tid = threadIdx.x;                      // 256 threads
  __shared__ float s1[256];
  __shared__ float s2[256];

  const float yi  = y[i % C];
  const float* row = logits + (size_t)i * N;

  // ---- row max ----
  float m = -3.402823466e+38f;
  for (int j = tid; j < N; j += 256) m = fmaxf(m, row[j]);
  s1[tid] = m; __syncthreads();
  for (int s = 128; s > 0; s >>= 1) {
    if (tid < s) s1[tid] = fmaxf(s1[tid], s1[tid + s]);
    __syncthreads();
  }
  m = s1[0]; __syncthreads();

  // ---- negative exp sum ----
  float sn = 0.0f;
  for (int j = tid; j < N; j += 256) {
    const float cj = y[j % C];
    const float e  = expf(row[j] - m);
    sn += (cj != yi) ? e : 0.0f;
  }
  s1[tid] = sn; __syncthreads();
  for (int s = 128; s > 0; s >>= 1) {
    if (tid < s) s1[tid] += s1[tid + s];
    __syncthreads();
  }
  sn = s1[0]; __syncthreads();

  // ---- positive log-prob sum ----
  float ps = 0.0f, cnt = 0.0f;
  for (int j = tid; j < N; j += 256) {
    const float cj = y[j % C];
    if (cj == yi && j != i) {
      const float lv = row[j] - m;
      ps  += lv - logf(expf(lv) + sn);
      cnt += 1.0f;
    }
  }
  s1[tid] = ps; s2[tid] = cnt; __syncthreads();
  for (int s = 128; s > 0; s >>= 1) {
    if (tid < s) { s1[tid] += s1[tid + s]; s2[tid] += s2[tid + s]; }
    __syncthreads();
  }
  if (tid == 0) rowvals[i] = -(TEMP / BASE_TEMP) * s1[0] / s2[0];
}

// ---------------------------------------------------------------------------
// Kernel 4: mean over rows -> scalar loss
// ---------------------------------------------------------------------------
__global__ void mean_rows(const float* __restrict__ rowvals,
                          float* __restrict__ out, int N) {
  __shared__ float s1[256];
  const int tid = threadIdx.x;
  float s = 0.0f;
  for (int j = tid; j < N; j += 256) s += rowvals[j];
  s1[tid] = s; __syncthreads();
  for (int st = 128; st > 0; st >>= 1) {
    if (tid < st) s1[tid] += s1[tid + st];
    __syncthreads();
  }
  if (tid == 0) out[0] = s1[0] / (float)N;
}

// ---------------------------------------------------------------------------
extern "C" void kernel_launch(void* const* d_in, const int* in_sizes, int n_in,
                              void* d_out, int out_size, void* d_ws, size_t ws_size,
                              hipStream_t stream) {
  const float* feats = (const float*)d_in[0];   // (B, D, L) f32
  const float* y     = (const float*)d_in[1];   // (C,)     f32
  const int*   sidx  = (const int*)d_in[2];     // (C, V)   i32

  const int C = in_sizes[1];
  const int V = in_sizes[2] / C;
  const int N = C * V;                          // 1024 for this config

  // Workspace layout: A (N*D f32) | logits (N*N f32) | rowvals (N f32)
  float* A       = (float*)d_ws;
  float* logits  = A + (size_t)N * D_DIM;
  float* rowvals = logits + (size_t)N * N;

  gather_anchor<<<N, D_DIM, 0, stream>>>(feats, sidx, A, C, V);

  dim3 gblock(32, 4);
  dim3 ggrid((N / 64 + 3) / 4, N / 16);
  wmma_gram<<<ggrid, gblock, 0, stream>>>(A, logits, N);

  row_reduce<<<N, 256, 0, stream>>>(logits, y, rowvals, N, C);

  mean_rows<<<1, 256, 0, stream>>>(rowvals, (float*)d_out, N);
}